// Motion_Exctractor_MEAN_3281355014529
// MI455X (gfx1250) — compile-verified
//
#include <hip/hip_runtime.h>
#include <math.h>

#define N_   16
#define C_   64
#define F_   8
#define HW_  3136
#define EMB_ 20
#define ITER_ 128
#define OUT_ 256
#define EPS_ 1e-5f
#define BIGF 3.402823466e38f

typedef __attribute__((ext_vector_type(2))) float v2f;
typedef __attribute__((ext_vector_type(8))) float v8f;

// ---- Kernel 1: single pass over h per (n,c,f) row:
//      async-LDS stage, row sum/sumsq partials, raw top-5 (desc) + bottom-5 (asc)
__global__ void __launch_bounds__(256) k_pass1(const float* __restrict__ h,
                                               float* __restrict__ rowpart,   // 8192 x 2
                                               float* __restrict__ rawsel) {  // 8192 x 10
  int row = blockIdx.x;                 // (n*C + c)*F + f ; flat base = row*HW
  __shared__ float buf[HW_] __attribute__((aligned(16)));
  __shared__ float rv[256];
  __shared__ float r2s[256];
  __shared__ int   ri[256];
  int tid = threadIdx.x;
  const float* p = h + (size_t)row * HW_;
  // --- async global -> LDS: 3 x (256 lanes x 16B) = 3072 floats ---
  unsigned ldsbase = (unsigned)(unsigned long long)(void*)buf;   // low 32b = LDS offset
  #pragma unroll
  for (int it = 0; it < 3; ++it) {
    int idx = (it * 256 + tid) * 4;                              // float index, 16B aligned
    unsigned loff = ldsbase + (unsigned)(idx * 4);
    unsigned long long ga = (unsigned long long)(const void*)(p + idx);
    asm volatile("global_load_async_to_lds_b128 %0, %1, off"
                 :: "v"(loff), "v"(ga) : "memory");
  }
  if (tid < HW_ - 3072) {
    __builtin_prefetch(p + 3072 + tid, 0, 1);                    // global_prefetch_b8
    buf[3072 + tid] = p[3072 + tid];                             // 64-float tail
  }
  asm volatile("s_wait_asynccnt 0x0" ::: "memory");
  __syncthreads();
  // --- row partial sums for BN stats ---
  {
    float s = 0.f, s2 = 0.f;
    for (int i = tid; i < HW_; i += 256) { float v = buf[i]; s += v; s2 += v * v; }
    rv[tid] = s; r2s[tid] = s2;
    __syncthreads();
    for (int st = 128; st > 0; st >>= 1) {
      if (tid < st) { rv[tid] += rv[tid + st]; r2s[tid] += r2s[tid + st]; }
      __syncthreads();
    }
    if (tid == 0) { rowpart[row * 2] = rv[0]; rowpart[row * 2 + 1] = r2s[0]; }
    __syncthreads();
  }
  // --- raw top-5 (descending), mask winners with -BIG ---
  for (int pass = 0; pass < 5; ++pass) {
    float best = -BIGF; int bi = 0;
    for (int i = tid; i < HW_; i += 256) { float v = buf[i]; if (v > best) { best = v; bi = i; } }
    rv[tid] = best; ri[tid] = bi;
    __syncthreads();
    for (int st = 128; st > 0; st >>= 1) {
      if (tid < st) {
        if (rv[tid + st] > rv[tid] || (rv[tid + st] == rv[tid] && ri[tid + st] < ri[tid])) {
          rv[tid] = rv[tid + st]; ri[tid] = ri[tid + st];
        }
      }
      __syncthreads();
    }
    if (tid == 0) { rawsel[row * 10 + pass] = rv[0]; buf[ri[0]] = -BIGF; }
    __syncthreads();
  }
  // --- raw bottom-5 (ascending); skip -BIG sentinels, mask winners with +BIG ---
  for (int pass = 0; pass < 5; ++pass) {
    float best = BIGF; int bi = 0;
    for (int i = tid; i < HW_; i += 256) {
      float v = buf[i];
      if (v < best && v > -1.0e38f) { best = v; bi = i; }
    }
    rv[tid] = best; ri[tid] = bi;
    __syncthreads();
    for (int st = 128; st > 0; st >>= 1) {
      if (tid < st) {
        if (rv[tid + st] < rv[tid] || (rv[tid + st] == rv[tid] && ri[tid + st] < ri[tid])) {
          rv[tid] = rv[tid + st]; ri[tid] = ri[tid + st];
        }
      }
      __syncthreads();
    }
    if (tid == 0) { rawsel[row * 10 + 5 + pass] = rv[0]; buf[ri[0]] = BIGF; }
    __syncthreads();
  }
}

// ---- Kernel 2: fold 128 row-partials per channel -> scale/shift ----
__global__ void __launch_bounds__(128) k_chanstats(const float* __restrict__ rowpart,
                                                   const float* __restrict__ g,
                                                   const float* __restrict__ b,
                                                   float* __restrict__ scsh) {  // 64 x 2
  int c = blockIdx.x;
  int t = threadIdx.x;                          // 0..127 -> (n,f)
  int n = t >> 3, f = t & 7;
  int row = n * (C_ * F_) + c * F_ + f;
  __shared__ float r1[128], r2[128];
  r1[t] = rowpart[row * 2]; r2[t] = rowpart[row * 2 + 1];
  __syncthreads();
  for (int st = 64; st > 0; st >>= 1) {
    if (t < st) { r1[t] += r1[t + st]; r2[t] += r2[t + st]; }
    __syncthreads();
  }
  if (t == 0) {
    const float cnt = (float)(N_ * F_ * HW_);
    float mean = r1[0] / cnt;
    float var = r2[0] / cnt - mean * mean;
    float scale = g[c] * rsqrtf(var + EPS_);
    scsh[c * 2] = scale; scsh[c * 2 + 1] = b[c] - mean * scale;
  }
}

// ---- Kernel 3: sign-aware top-5 selection + attention chain -> centroid (N,C,F,2) ----
__global__ void __launch_bounds__(256) k_att1(const float* __restrict__ rawsel,
                                              const float* __restrict__ scsh,
                                              const float* __restrict__ up2_w, const float* __restrict__ up2_b,
                                              const float* __restrict__ up3_w, const float* __restrict__ up3_b,
                                              const float* __restrict__ wW, const float* __restrict__ wproj,
                                              float* __restrict__ cen) {
  __shared__ float s_up2w[125], s_up2b[25], s_up3w[125], s_up3b[5], s_wW[625], s_wproj[125];
  int tid = threadIdx.x;
  for (int i = tid; i < 125; i += 256) { s_up2w[i] = up2_w[i]; s_up3w[i] = up3_w[i]; s_wproj[i] = wproj[i]; }
  for (int i = tid; i < 625; i += 256) s_wW[i] = wW[i];
  if (tid < 25) s_up2b[tid] = up2_b[tid];
  if (tid < 5)  s_up3b[tid] = up3_b[tid];
  __syncthreads();
  int row = blockIdx.x * 256 + tid;     // exactly 8192 threads
  int c = (row >> 3) & (C_ - 1);        // (row / F_) % C_
  float scale = scsh[c * 2], shift = scsh[c * 2 + 1];
  float tv[5];
  if (scale >= 0.f) {
    for (int k = 0; k < 5; ++k) tv[k] = rawsel[row * 10 + k] * scale + shift;       // affine(top5)
  } else {
    for (int k = 0; k < 5; ++k) tv[k] = rawsel[row * 10 + 9 - k] * scale + shift;   // affine(rev(bot5))
  }
  float t[25];
  for (int j = 0; j < 25; ++j) {
    float a = s_up2b[j];
    for (int k = 0; k < 5; ++k) a += tv[k] * s_up2w[j * 5 + k];
    t[j] = a;
  }
  float u[25];
  for (int i = 0; i < 25; ++i) {
    float a = 0.f;
    for (int j = 0; j < 25; ++j) a += t[j] * s_wW[j * 25 + i];
    u[i] = tanhf(a);
  }
  float z[5]; float zm = -BIGF;
  for (int k = 0; k < 5; ++k) {
    float a = 0.f;
    for (int i = 0; i < 25; ++i) a += u[i] * s_wproj[i * 5 + k];
    z[k] = a; zm = fmaxf(zm, a);
  }
  float se = 0.f;
  for (int k = 0; k < 5; ++k) { z[k] = expf(z[k] - zm); se += z[k]; }
  float t3[5];
  for (int k = 0; k < 5; ++k) {
    float a = s_up3b[k];
    for (int j = 0; j < 25; ++j) a += t[j] * s_up3w[k * 25 + j];
    t3[k] = a * (z[k] / se);            // t3 * att fused
  }
  float c0 = 0.f, c1 = 0.f;
  for (int j = 0; j < 25; ++j) {
    float sj = s_up2b[j];
    for (int k = 0; k < 5; ++k) sj += t3[k] * s_up2w[j * 5 + k];
    c0 += sj * (float)(j / 5);
    c1 += sj * (float)(j % 5);
  }
  cen[row * 2] = c0; cen[row * 2 + 1] = c1;
}

// ---- Kernel 4: temporal diff + attention -> zero-padded si (N,C,7,22) ----
__global__ void __launch_bounds__(64) k_att2(const float* __restrict__ cen,
                                             const float* __restrict__ up_w, const float* __restrict__ up_b,
                                             const float* __restrict__ wW2, const float* __restrict__ wproj2,
                                             float* __restrict__ sip) {
  int r = blockIdx.x * 64 + threadIdx.x;     // r = n*C + c, 1024 total
  if (r >= N_ * C_) return;
  float d0[7], d1[7];
  for (int f = 0; f < 7; ++f) {
    d0[f] = cen[(r * F_ + f + 1) * 2]     - cen[(r * F_ + f) * 2];
    d1[f] = cen[(r * F_ + f + 1) * 2 + 1] - cen[(r * F_ + f) * 2 + 1];
  }
  float c2[7][EMB_];
  for (int f = 0; f < 7; ++f)
    for (int e = 0; e < EMB_; ++e)
      c2[f][e] = d0[f] * up_w[e * 2] + d1[f] * up_w[e * 2 + 1] + up_b[e];
  float zz[7]; float zm = -BIGF;
  for (int i = 0; i < 7; ++i) {
    float a = 0.f;
    for (int e = 0; e < EMB_; ++e) {
      float s = 0.f;
      for (int f = 0; f < 7; ++f) s += wW2[i * 7 + f] * c2[f][e];
      a += tanhf(s) * wproj2[e];
    }
    zz[i] = a; zm = fmaxf(zm, a);
  }
  float se = 0.f;
  for (int i = 0; i < 7; ++i) { zz[i] = expf(zz[i] - zm); se += zz[i]; }
  for (int f = 0; f < 7; ++f) {
    float a2 = zz[f] / se;
    int rowb = (r * 7 + f) * 22;
    sip[rowb] = 0.f; sip[rowb + 21] = 0.f;            // zero pad for conv halo
    for (int e = 0; e < EMB_; ++e) sip[rowb + 1 + e] = c2[f][e] * a2;
  }
}

// ---- Weight repack: 3 taps -> 4 taps (tap3 = 0) ----
__global__ void __launch_bounds__(256) k_repack_w(const float* __restrict__ wsrc,
                                                  float* __restrict__ wdst,
                                                  int nelems, int cin) {
  int idx = blockIdx.x * 256 + threadIdx.x;          // over O * cin * 4
  if (idx >= nelems) return;
  int t = idx & 3;
  int oc = idx >> 2;                                 // o*cin + c
  int c = oc % cin, o = oc / cin;
  wdst[idx] = (t < 3) ? wsrc[(o * cin + c) * 3 + t] : 0.f;
}

// ---- Kernel 5: conv1 implicit GEMM via V_WMMA_F32_16X16X4_F32 ----
// A = w1r (128 x 256), B = padded im2col of sip, tile grid 8 x 70 over (128 x 1120)
__global__ void __launch_bounds__(32) k_conv1_wmma(const float* __restrict__ sip,
                                                   const float* __restrict__ w1r,
                                                   float* __restrict__ out1) {
  int tile = blockIdx.x;
  int mt = tile / 70, nt = tile % 70;
  int m0 = mt * 16, n0 = nt * 16;
  int lane = threadIdx.x;
  int half = lane >> 4;                  // lanes 16-31 hold K+2 / M+8
  int l15 = lane & 15;
  int col = n0 + l15;                    // output column (n,f,x)
  int n = col / 70; int rem = col % 70; int f = rem / 10; int x = rem % 10;
  const float* ap = w1r + (m0 + l15) * 256 + 2 * half;            // stride 4 per c
  const float* bp = sip + (n * 448 + f) * 22 + 2 * x + 2 * half;  // stride 154 per c
  v8f acc = {};
  #pragma unroll 4
  for (int c = 0; c < 64; ++c) {
    v2f a = *(const v2f*)(ap + c * 4);
    v2f bb = *(const v2f*)(bp + c * 154);
    acc = __builtin_amdgcn_wmma_f32_16x16x4_f32(false, a, false, bb, (short)0, acc, false, false);
  }
  int mrow = m0 + half * 8;
  #pragma unroll
  for (int rr = 0; rr < 8; ++rr) out1[(mrow + rr) * 1120 + col] = acc[rr];
}

// ---- Kernel 6: BN1 + tanh, write zero-padded (128 x 112 x 12) ----
__global__ void __launch_bounds__(256) k_bn1_pad(const float* __restrict__ out1,
                                                 const float* __restrict__ g,
                                                 const float* __restrict__ b,
                                                 float* __restrict__ out1p) {
  int o = blockIdx.x;
  const float* p = out1 + (size_t)o * 1120;
  float s = 0.f, s2 = 0.f;
  for (int i = threadIdx.x; i < 1120; i += 256) { float v = p[i]; s += v; s2 += v * v; }
  __shared__ float r1[256], r2[256];
  r1[threadIdx.x] = s; r2[threadIdx.x] = s2;
  __syncthreads();
  for (int st = 128; st > 0; st >>= 1) {
    if (threadIdx.x < st) { r1[threadIdx.x] += r1[threadIdx.x + st]; r2[threadIdx.x] += r2[threadIdx.x + st]; }
    __syncthreads();
  }
  float mean = r1[0] / 1120.f;
  float var = r2[0] / 1120.f - mean * mean;
  float scale = g[o] * rsqrtf(var + EPS_);
  float shift = b[o] - mean * scale;
  float* q = out1p + (size_t)o * 1344;
  for (int i = threadIdx.x; i < 1120; i += 256) {
    int gg = i / 10, x = i % 10;
    q[gg * 12 + 1 + x] = tanhf(p[i] * scale + shift);
  }
  for (int gg = threadIdx.x; gg < 112; gg += 256) { q[gg * 12] = 0.f; q[gg * 12 + 11] = 0.f; }
}

// ---- Kernel 7: conv2 implicit GEMM via WMMA ----
// A = w2r (256 x 512), B = padded im2col of out1p, tile grid 16 x 35 over (256 x 560)
__global__ void __launch_bounds__(32) k_conv2_wmma(const float* __restrict__ x1p,
                                                   const float* __restrict__ w2r,
                                                   float* __restrict__ out2) {
  int tile = blockIdx.x;
  int mt = tile / 35, nt = tile % 35;
  int m0 = mt * 16, n0 = nt * 16;
  int lane = threadIdx.x;
  int half = lane >> 4;
  int l15 = lane & 15;
  int col = n0 + l15;
  int n = col / 35; int rem = col % 35; int f = rem / 5; int x2 = rem % 5;
  const float* ap = w2r + (m0 + l15) * 512 + 2 * half;           // stride 4 per c
  const float* bp = x1p + (n * 7 + f) * 12 + 2 * x2 + 2 * half;  // stride 1344 per c
  v8f acc = {};
  #pragma unroll 4
  for (int c = 0; c < 128; ++c) {
    v2f a = *(const v2f*)(ap + c * 4);
    v2f bb = *(const v2f*)(bp + c * 1344);
    acc = __builtin_amdgcn_wmma_f32_16x16x4_f32(false, a, false, bb, (short)0, acc, false, false);
  }
  int mrow = m0 + half * 8;
  #pragma unroll
  for (int rr = 0; rr < 8; ++rr) out2[(mrow + rr) * 560 + col] = acc[rr];
}

// ---- Kernel 8: BN2 + tanh + mean over width -> output (N,256,7,1) ----
__global__ void __launch_bounds__(256) k_bn2_out(const float* __restrict__ x2,
                                                 const float* __restrict__ g,
                                                 const float* __restrict__ b,
                                                 float* __restrict__ out) {
  int o = blockIdx.x;                    // 256 channels
  const float* p = x2 + (size_t)o * 560;
  float s = 0.f, s2 = 0.f;
  for (int i = threadIdx.x; i < 560; i += 256) { float v = p[i]; s += v; s2 += v * v; }
  __shared__ float r1[256], r2[256];
  r1[threadIdx.x] = s; r2[threadIdx.x] = s2;
  __syncthreads();
  for (int st = 128; st > 0; st >>= 1) {
    if (threadIdx.x < st) { r1[threadIdx.x] += r1[threadIdx.x + st]; r2[threadIdx.x] += r2[threadIdx.x + st]; }
    __syncthreads();
  }
  float mean = r1[0] / 560.f;
  float var = r2[0] / 560.f - mean * mean;
  float scale = g[o] * rsqrtf(var + EPS_);
  float shift = b[o] - mean * scale;
  if (threadIdx.x < 112) {               // 16*7 (n,f) groups
    int n = threadIdx.x / 7, f = threadIdx.x % 7;
    float a = 0.f;
    for (int x = 0; x < 5; ++x) a += tanhf(p[threadIdx.x * 5 + x] * scale + shift);
    out[(n * OUT_ + o) * 7 + f] = a * 0.2f;
  }
}

// ---- Launch ----
extern "C" void kernel_launch(void* const* d_in, const int* in_sizes, int n_in,
                              void* d_out, int out_size, void* d_ws, size_t ws_size,
                              hipStream_t stream) {
  const float* h      = (const float*)d_in[0];
  const float* bn0_g  = (const float*)d_in[1];
  const float* bn0_b  = (const float*)d_in[2];
  const float* up_w   = (const float*)d_in[3];
  const float* up_b   = (const float*)d_in[4];
  const float* up2_w  = (const float*)d_in[5];
  const float* up2_b  = (const float*)d_in[6];
  const float* up3_w  = (const float*)d_in[7];
  const float* up3_b  = (const float*)d_in[8];
  const float* wW     = (const float*)d_in[9];
  const float* wproj  = (const float*)d_in[10];
  const float* wW2    = (const float*)d_in[11];
  const float* wproj2 = (const float*)d_in[12];
  const float* c1w    = (const float*)d_in[13];
  const float* bn1_g  = (const float*)d_in[14];
  const float* bn1_b  = (const float*)d_in[15];
  const float* c2w    = (const float*)d_in[16];
  const float* bn2_g  = (const float*)d_in[17];
  const float* bn2_b  = (const float*)d_in[18];
  (void)in_sizes; (void)n_in; (void)out_size; (void)ws_size;

  float* ws      = (float*)d_ws;
  float* rowpart = ws;                  // 8192*2   = 16384
  float* rawsel  = rowpart + 16384;     // 8192*10  = 81920
  float* scsh    = rawsel  + 81920;     // 64*2     = 128
  float* cen     = scsh    + 128;       // 8192*2   = 16384
  float* sip     = cen     + 16384;     // 16*64*7*22 = 157696
  float* w1r     = sip     + 157696;    // 128*256  = 32768
  float* w2r     = w1r     + 32768;     // 256*512  = 131072
  float* out1    = w2r     + 131072;    // 128*1120 = 143360
  float* out1p   = out1    + 143360;    // 128*1344 = 172032
  float* out2    = out1p   + 172032;    // 256*560  = 143360

  k_pass1<<<N_ * C_ * F_, 256, 0, stream>>>(h, rowpart, rawsel);
  k_chanstats<<<C_, 128, 0, stream>>>(rowpart, bn0_g, bn0_b, scsh);
  k_repack_w<<<(ITER_ * 256 + 255) / 256, 256, 0, stream>>>(c1w, w1r, ITER_ * 256, C_);
  k_repack_w<<<(OUT_ * 512 + 255) / 256, 256, 0, stream>>>(c2w, w2r, OUT_ * 512, ITER_);
  k_att1<<<32, 256, 0, stream>>>(rawsel, scsh, up2_w, up2_b, up3_w, up3_b, wW, wproj, cen);
  k_att2<<<16, 64, 0, stream>>>(cen, up_w, up_b, wW2, wproj2, sip);
  k_conv1_wmma<<<8 * 70, 32, 0, stream>>>(sip, w1r, out1);
  k_bn1_pad<<<ITER_, 256, 0, stream>>>(out1, bn1_g, bn1_b, out1p);
  k_conv2_wmma<<<16 * 35, 32, 0, stream>>>(out1p, w2r, out2);
  k_bn2_out<<<OUT_, 256, 0, stream>>>(out2, bn2_g, bn2_b, (float*)d_out);
}